// ConditionalDDPM_49452253446979
// MI455X (gfx1250) — compile-verified
//
#include <hip/hip_runtime.h>

typedef float v4f __attribute__((ext_vector_type(4)));
typedef float v2f __attribute__((ext_vector_type(2)));

#define DDPM_T   1000
#define DDPM_B1  1e-4f
#define DDPM_BT  2e-2f

// ---------------------------------------------------------------------------
// Kernel 1: build { sqrt(alpha_bar[k]), sqrt(1-alpha_bar[k]) } table (8 KB).
// Sequential cumprod exactly mirrors jnp.cumprod(1 - betas).
// ---------------------------------------------------------------------------
__global__ void ddpm_table_kernel(float* __restrict__ table) {
    if (blockIdx.x == 0 && threadIdx.x == 0) {
        const float slope = (DDPM_BT - DDPM_B1) / (float)(DDPM_T - 1);
        float ab = 1.0f;
        for (int k = 0; k < DDPM_T; ++k) {
            const float beta = DDPM_B1 + slope * (float)k;
            ab *= (1.0f - beta);
            table[2 * k + 0] = sqrtf(ab);
            table[2 * k + 1] = sqrtf(1.0f - ab);
        }
    }
}

// ---------------------------------------------------------------------------
// Kernel 2: streaming mix. Per wave: double-buffered async DMA of the two
// input streams into LDS (ASYNCcnt, non-temporal policy), FMA, non-temporal
// b128 store. LDS layout: lbuf[stage][stream][wave][lane] -> stage stride
// 8192 B, stream stride 4096 B. 16 KB per 256-thread block.
// ---------------------------------------------------------------------------
__global__ __launch_bounds__(256) void ddpm_mix_kernel(
    const float* __restrict__ img, const float* __restrict__ e,
    const int* __restrict__ t, const float* __restrict__ table,
    float* __restrict__ out, unsigned ntiles)
{
    __shared__ v4f lbuf[2][2][8][32];   // 16 KB

    const unsigned lane = threadIdx.x & 31u;
    const unsigned wib  = threadIdx.x >> 5;      // wave index in block
    const unsigned wid  = blockIdx.x * 8u + wib; // global wave id
    const unsigned W    = gridDim.x * 8u;        // total waves

    // Low 32 bits of a generic pointer into LDS == LDS byte offset.
    const unsigned lbase =
        (unsigned)(unsigned long long)&lbuf[0][0][wib][lane];

    const v4f* img4 = (const v4f*)img;
    const v4f* e4   = (const v4f*)e;
    v4f*       out4 = (v4f*)out;
    const v2f* tab2 = (const v2f*)table;

    unsigned tile  = wid;
    unsigned stage = 0;

    if (tile < ntiles) {
        const unsigned i4 = tile * 32u + lane;
        unsigned long long gi = (unsigned long long)(img4 + i4);
        unsigned long long ge = (unsigned long long)(e4 + i4);
        unsigned la = lbase;            // stage 0, stream img
        unsigned lb = lbase + 4096u;    // stage 0, stream e
        asm volatile(
            "global_load_async_to_lds_b128 %0, %2, off th:TH_LOAD_NT\n\t"
            "global_load_async_to_lds_b128 %1, %3, off th:TH_LOAD_NT"
            :: "v"(la), "v"(lb), "v"(gi), "v"(ge) : "memory");
    }

    while (tile < ntiles) {
        const unsigned next = tile + W;   // wave-uniform control flow

        if (next < ntiles) {
            // Prefetch next tile into the other stage, then wait for the
            // older pair (2 newest async loads may remain outstanding).
            const unsigned i4n = next * 32u + lane;
            unsigned long long gi = (unsigned long long)(img4 + i4n);
            unsigned long long ge = (unsigned long long)(e4 + i4n);
            const unsigned soff = (stage ^ 1u) * 8192u;
            unsigned la = lbase + soff;
            unsigned lb = lbase + soff + 4096u;
            asm volatile(
                "global_load_async_to_lds_b128 %0, %2, off th:TH_LOAD_NT\n\t"
                "global_load_async_to_lds_b128 %1, %3, off th:TH_LOAD_NT"
                :: "v"(la), "v"(lb), "v"(gi), "v"(ge) : "memory");
            asm volatile("s_wait_asynccnt 0x2" ::: "memory");
        } else {
            asm volatile("s_wait_asynccnt 0x0" ::: "memory");
        }

        const unsigned i4 = tile * 32u + lane;
        const v4f vi = lbuf[stage][0][wib][lane];   // ds_load_b128
        const v4f ve = lbuf[stage][1][wib][lane];   // ds_load_b128

        const unsigned imgIdx = i4 / 196u;          // 784/4 f32x4 per image
        const int tt = t[imgIdx];                   // cached (L2-resident)
        const v2f s  = tab2[tt];                    // cached 8 KB table

        v4f o;
        o.x = s.x * vi.x + s.y * ve.x;
        o.y = s.x * vi.y + s.y * ve.y;
        o.z = s.x * vi.z + s.y * ve.z;
        o.w = s.x * vi.w + s.y * ve.w;
        __builtin_nontemporal_store(o, out4 + i4);  // streaming, bypass reuse

        stage ^= 1u;
        tile = next;
    }
}

// ---------------------------------------------------------------------------
// Scalar tail (not launched for the reference shape; kept for generality).
// ---------------------------------------------------------------------------
__global__ void ddpm_tail_kernel(const float* __restrict__ img,
                                 const float* __restrict__ e,
                                 const int* __restrict__ t,
                                 const float* __restrict__ table,
                                 float* __restrict__ out,
                                 long long start, long long n)
{
    long long i = start + (long long)blockIdx.x * blockDim.x + threadIdx.x;
    if (i < n) {
        int im = (int)(i / 784);
        int tt = t[im];
        float sa = table[2 * tt + 0];
        float sb = table[2 * tt + 1];
        out[i] = sa * img[i] + sb * e[i];
    }
}

extern "C" void kernel_launch(void* const* d_in, const int* in_sizes, int n_in,
                              void* d_out, int out_size, void* d_ws, size_t ws_size,
                              hipStream_t stream) {
    const float* img = (const float*)d_in[0];
    const float* e   = (const float*)d_in[1];
    const int*   t   = (const int*)d_in[2];
    float*       out = (float*)d_out;
    float*       table = (float*)d_ws;   // 2 * 1000 floats = 8 KB

    const long long n  = (long long)in_sizes[0];   // 51,380,224
    const long long n4 = n / 4;                    // 12,845,056
    const unsigned ntiles = (unsigned)(n4 / 32);   // 401,408 wave-tiles

    ddpm_table_kernel<<<1, 64, 0, stream>>>(table);
    ddpm_mix_kernel<<<4096, 256, 0, stream>>>(img, e, t, table, out, ntiles);

    const long long covered = (long long)ntiles * 128;
    if (covered < n) {
        const long long rem = n - covered;
        const int blocks = (int)((rem + 255) / 256);
        ddpm_tail_kernel<<<blocks, 256, 0, stream>>>(img, e, t, table, out,
                                                     covered, n);
    }
}